// SobelLossV2_4818953306360
// MI455X (gfx1250) — compile-verified
//
#include <hip/hip_runtime.h>
#include <hip/hip_bf16.h>

// Problem constants (fixed by the reference: [16,3,512,512] f32)
#define IMG_H   512
#define IMG_W   512
#define IMG_B   16
#define HW      (IMG_H * IMG_W)            // 262144
#define N_GRAY  (IMG_B * HW)               // 4,194,304
#define N_PIX   (IMG_B * 3 * HW)           // 12,582,912

#define RED_BLOCKS 2048
#define BLOCK      256                      // 8 waves (wave32)
#define TS         32                       // stencil tile (512 % 32 == 0)

typedef float        v2f __attribute__((ext_vector_type(2)));
typedef float        v8f __attribute__((ext_vector_type(8)));
typedef unsigned int v4u __attribute__((ext_vector_type(4)));
typedef int          v8i __attribute__((ext_vector_type(8)));
typedef int          v4i __attribute__((ext_vector_type(4)));

// ---------------------------------------------------------------------------
// Tensor Data Mover: 2D tile load (global -> LDS, packed row-major in LDS).
// D# packed per CDNA5 ISA. Tiles are always fully
// interior here (512 % 32 == 0), so OOB/padding features are unused.
// Issued once per workgroup by wave 0; completion via s_wait_tensorcnt.
// ---------------------------------------------------------------------------
__device__ __forceinline__ void tdm_load_tile_2d(unsigned lds_byte_off,
                                                 const float* gsrc,
                                                 unsigned tile_w, unsigned tile_h,
                                                 unsigned rem_w, unsigned rem_h,
                                                 unsigned stride_elems) {
    const unsigned long long ga = (unsigned long long)(uintptr_t)gsrc;
    v4u g0;
    g0.x = 1u;                                                // count=1, user mode
    g0.y = lds_byte_off;                                      // lds_addr (bytes)
    g0.z = (unsigned)(ga & 0xFFFFFFFFu);                      // global_addr[31:0]
    g0.w = (unsigned)((ga >> 32) & 0x01FFFFFFu) | (2u << 30); // addr[56:32] | type=2
    v8i g1;
    g1[0] = (int)(2u << 16);                                  // data_size = 4 bytes
    g1[1] = (int)((rem_w & 0xFFFFu) << 16);                   // tensor_dim0[15:0]
    g1[2] = (int)((rem_w >> 16) | ((rem_h & 0xFFFFu) << 16)); // dim0 hi | dim1 lo
    g1[3] = (int)((rem_h >> 16) | (tile_w << 16));            // dim1 hi | tile_dim0
    g1[4] = (int)(tile_h & 0xFFFFu);                          // tile_dim1 (tile_dim2=0)
    g1[5] = (int)stride_elems;                                // tensor_dim0_stride lo
    g1[6] = 0;
    g1[7] = 0;
    const v4i gz = {0, 0, 0, 0};
#if __has_include(<hip/amd_detail/amd_gfx1250_TDM.h>)
    const v8i gz8 = {0, 0, 0, 0, 0, 0, 0, 0};
    __builtin_amdgcn_tensor_load_to_lds(g0, g1, gz, gz, gz8, 0);
#else
    __builtin_amdgcn_tensor_load_to_lds(g0, g1, gz, gz, 0);
#endif
}

// ---------------------------------------------------------------------------
// Wave-level f32 sum using V_WMMA_F32_16X16X4_F32 (exact f32 accumulate).
// A = ones(16x4); B carries one partial per lane. Every row of D holds the
// 16 column sums, so D VGPR0 duplicates them across the wave: summing it over
// 32 lanes gives 2 * total. Requires EXEC all-ones (call sites are uniform).
// ---------------------------------------------------------------------------
__device__ __forceinline__ float wave_reduce_f32(float v) {
    v2f a; a[0] = 1.0f; a[1] = 1.0f;
    v2f b; b[0] = v;    b[1] = 0.0f;
    v8f c = {};
    c = __builtin_amdgcn_wmma_f32_16x16x4_f32(false, a, false, b,
                                              (short)0, c, false, false);
    float s = c[0];
    s += __shfl_xor(s, 16, 32);
    s += __shfl_xor(s,  8, 32);
    s += __shfl_xor(s,  4, 32);
    s += __shfl_xor(s,  2, 32);
    s += __shfl_xor(s,  1, 32);
    return 0.5f * s;
}

// ---------------------------------------------------------------------------
// Pass 1: fused grayscale + L1 partial sums (reads the full 100 MB once).
//   acc[0] += sum |pred - target| ;  acc[1] += sum |pred_gray - target_gray|
//   gray[i] = target_gray
// ---------------------------------------------------------------------------
__global__ void k_pass1(const float* __restrict__ pred,
                        const float* __restrict__ tgt,
                        float* __restrict__ gray,
                        float* __restrict__ acc) {
    __shared__ float s_acc[2];
    if (threadIdx.x == 0) { s_acc[0] = 0.0f; s_acc[1] = 0.0f; }
    __syncthreads();

    const int tid    = blockIdx.x * BLOCK + threadIdx.x;
    const int stride = gridDim.x * BLOCK;   // N_GRAY % stride == 0 -> uniform trips

    float psum = 0.0f, gsum = 0.0f;
    for (int i = tid; i < N_GRAY; i += stride) {
        const int b   = i >> 18;
        const int pix = i & (HW - 1);
        const float* p = pred + (size_t)b * 3 * HW + pix;
        const float* t = tgt  + (size_t)b * 3 * HW + pix;
        const float pr = p[0], pg = p[HW], pb = p[2 * HW];
        const float tr = t[0], tg = t[HW], tb = t[2 * HW];
        psum += fabsf(pr - tr) + fabsf(pg - tg) + fabsf(pb - tb);
        const float pgray = 0.299f * pr + 0.587f * pg + 0.114f * pb;
        const float tgray = 0.299f * tr + 0.587f * tg + 0.114f * tb;
        gsum += fabsf(pgray - tgray);
        gray[i] = tgray;
    }

    const float wp = wave_reduce_f32(psum);
    const float wg = wave_reduce_f32(gsum);
    if ((threadIdx.x & 31) == 0) { atomicAdd(&s_acc[0], wp); atomicAdd(&s_acc[1], wg); }
    __syncthreads();
    if (threadIdx.x == 0) { atomicAdd(&acc[0], s_acc[0]); atomicAdd(&acc[1], s_acc[1]); }
}

// ---------------------------------------------------------------------------
// Zero-padded gray access + Sobel magnitude (conv semantics).
// ---------------------------------------------------------------------------
__device__ __forceinline__ float gray_at(const float* __restrict__ g, int h, int w) {
    if (h < 0 || h >= IMG_H || w < 0 || w >= IMG_W) return 0.0f;
    return g[h * IMG_W + w];
}

__device__ __forceinline__ float sobel_mag(const float* __restrict__ g, int h, int w) {
    const float a00 = gray_at(g, h - 1, w - 1), a01 = gray_at(g, h - 1, w), a02 = gray_at(g, h - 1, w + 1);
    const float a10 = gray_at(g, h,     w - 1),                               a12 = gray_at(g, h,     w + 1);
    const float a20 = gray_at(g, h + 1, w - 1), a21 = gray_at(g, h + 1, w), a22 = gray_at(g, h + 1, w + 1);
    const float gx = (a02 - a00) + 2.0f * (a12 - a10) + (a22 - a20);
    const float gy = (a20 - a00) + 2.0f * (a21 - a01) + (a22 - a02);
    return sqrtf(gx * gx + gy * gy);
}

// ---------------------------------------------------------------------------
// Pass 2: Sobel magnitude. TDM stages each 32x32 gray tile into LDS; interior
// stencil reads hit LDS, halo ring falls back to L2 with zero padding.
// Grid: (16,16,16) tiles; block 256 = 32x8, 4 rows per thread.
// ---------------------------------------------------------------------------
__global__ void k_mag(const float* __restrict__ gray, float* __restrict__ mag) {
    __shared__ float tile[TS * TS];
    const int b  = blockIdx.z;
    const int h0 = blockIdx.y * TS;
    const int w0 = blockIdx.x * TS;
    const float* g = gray + (size_t)b * HW;

    if ((threadIdx.x >> 5) == 0) {          // wave 0 issues one TDM op per block
        tdm_load_tile_2d((unsigned)(uintptr_t)&tile[0],
                         g + h0 * IMG_W + w0, TS, TS,
                         (unsigned)(IMG_W - w0), (unsigned)(IMG_H - h0), IMG_W);
        __builtin_amdgcn_s_wait_tensorcnt(0);
    }
    __syncthreads();

    const int tx = threadIdx.x & 31;
    const int ty = threadIdx.x >> 5;
    float* mdst = mag + (size_t)b * HW;

    #pragma unroll
    for (int sub = 0; sub < 4; ++sub) {
        const int r = ty + sub * 8;
        auto val = [&](int rr, int cc) -> float {
            if (rr >= 0 && rr < TS && cc >= 0 && cc < TS) return tile[rr * TS + cc];
            return gray_at(g, h0 + rr, w0 + cc);        // halo: L2, zero-padded
        };
        const float a00 = val(r - 1, tx - 1), a01 = val(r - 1, tx), a02 = val(r - 1, tx + 1);
        const float a10 = val(r,     tx - 1),                        a12 = val(r,     tx + 1);
        const float a20 = val(r + 1, tx - 1), a21 = val(r + 1, tx), a22 = val(r + 1, tx + 1);
        const float gx = (a02 - a00) + 2.0f * (a12 - a10) + (a22 - a20);
        const float gy = (a20 - a00) + 2.0f * (a21 - a01) + (a22 - a02);
        mdst[(h0 + r) * IMG_W + (w0 + tx)] = sqrtf(gx * gx + gy * gy);
    }
}

// ---------------------------------------------------------------------------
// Pass 3: NMS with jnp.roll wrap-around neighbors; acc[2] += sum(edge_map).
// Same TDM tiling; halo ring uses wrapped global indices (L2).
// ---------------------------------------------------------------------------
__global__ void k_edge(const float* __restrict__ mag, float* __restrict__ acc) {
    __shared__ float tile[TS * TS];
    __shared__ float s_acc;
    if (threadIdx.x == 0) s_acc = 0.0f;

    const int b  = blockIdx.z;
    const int h0 = blockIdx.y * TS;
    const int w0 = blockIdx.x * TS;
    const float* m = mag + (size_t)b * HW;

    if ((threadIdx.x >> 5) == 0) {
        tdm_load_tile_2d((unsigned)(uintptr_t)&tile[0],
                         m + h0 * IMG_W + w0, TS, TS,
                         (unsigned)(IMG_W - w0), (unsigned)(IMG_H - h0), IMG_W);
        __builtin_amdgcn_s_wait_tensorcnt(0);
    }
    __syncthreads();

    const int tx = threadIdx.x & 31;
    const int ty = threadIdx.x >> 5;

    float esum = 0.0f;
    #pragma unroll
    for (int sub = 0; sub < 4; ++sub) {
        const int r = ty + sub * 8;
        auto val = [&](int rr, int cc) -> float {
            if (rr >= 0 && rr < TS && cc >= 0 && cc < TS) return tile[rr * TS + cc];
            const int hh = (h0 + rr + IMG_H) & (IMG_H - 1);   // wrap (jnp.roll)
            const int ww = (w0 + cc + IMG_W) & (IMG_W - 1);
            return m[hh * IMG_W + ww];
        };
        const float c   = tile[r * TS + tx];
        const float nU  = val(r - 1, tx    ), nD  = val(r + 1, tx    );
        const float nL  = val(r,     tx - 1), nR  = val(r,     tx + 1);
        const float nUL = val(r - 1, tx - 1), nDR = val(r + 1, tx + 1);
        const float nDL = val(r + 1, tx - 1), nUR = val(r - 1, tx + 1);
        const bool keep = (c > nU  && c > nD ) ||   // s=0: (dx,dy)=(0,1)
                          (c > nUL && c > nDR) ||   // s=1: (1,1)
                          (c > nL  && c > nR ) ||   // s=2: (1,0)
                          (c > nDL && c > nUR);     // s=3: (1,-1)
        esum += keep ? c : 0.0f;
    }

    const float we = wave_reduce_f32(esum);         // uniform flow, EXEC all-ones
    if ((threadIdx.x & 31) == 0) atomicAdd(&s_acc, we);
    __syncthreads();
    if (threadIdx.x == 0) atomicAdd(&acc[2], s_acc);
}

// Fallback (small ws): NMS with on-the-fly Sobel recompute from gray.
__global__ void k_edge_otf(const float* __restrict__ gray, float* __restrict__ acc) {
    __shared__ float s_acc;
    if (threadIdx.x == 0) s_acc = 0.0f;
    __syncthreads();

    const int tid    = blockIdx.x * BLOCK + threadIdx.x;
    const int stride = gridDim.x * BLOCK;

    float esum = 0.0f;
    for (int i = tid; i < N_GRAY; i += stride) {
        const int b   = i >> 18;
        const int pix = i & (HW - 1);
        const int h   = pix >> 9;
        const int w   = pix & (IMG_W - 1);
        const float* g = gray + (size_t)b * HW;

        const int hu = (h + IMG_H - 1) & (IMG_H - 1);
        const int hd = (h + 1)          & (IMG_H - 1);
        const int wl = (w + IMG_W - 1) & (IMG_W - 1);
        const int wr = (w + 1)          & (IMG_W - 1);

        const float c   = sobel_mag(g, h,  w );
        const float nU  = sobel_mag(g, hu, w ), nD  = sobel_mag(g, hd, w );
        const float nL  = sobel_mag(g, h,  wl), nR  = sobel_mag(g, h,  wr);
        const float nUL = sobel_mag(g, hu, wl), nDR = sobel_mag(g, hd, wr);
        const float nDL = sobel_mag(g, hd, wl), nUR = sobel_mag(g, hu, wr);

        const bool keep = (c > nU  && c > nD ) || (c > nUL && c > nDR) ||
                          (c > nL  && c > nR ) || (c > nDL && c > nUR);
        esum += keep ? c : 0.0f;
    }

    const float we = wave_reduce_f32(esum);
    if ((threadIdx.x & 31) == 0) atomicAdd(&s_acc, we);
    __syncthreads();
    if (threadIdx.x == 0) atomicAdd(&acc[2], s_acc);
}

// Final scalar: pixel_loss + mean(edge_map) * mae_gray
__global__ void k_final(const float* __restrict__ acc, float* __restrict__ out) {
    if (blockIdx.x == 0 && threadIdx.x == 0) {
        const float pixel_loss = acc[0] / (float)N_PIX;
        const float mae_gray   = acc[1] / (float)N_GRAY;
        const float edge_mean  = acc[2] / (float)N_GRAY;
        out[0] = pixel_loss + edge_mean * mae_gray;
    }
}

extern "C" void kernel_launch(void* const* d_in, const int* in_sizes, int n_in,
                              void* d_out, int out_size, void* d_ws, size_t ws_size,
                              hipStream_t stream) {
    const float* pred = (const float*)d_in[0];
    const float* tgt  = (const float*)d_in[1];
    float* out = (float*)d_out;

    char* ws = (char*)d_ws;
    const size_t GRAY_BYTES = (size_t)N_GRAY * sizeof(float);   // 16.8 MB
    float* gray = (float*)ws;

    const bool have_mag = ws_size >= 2 * GRAY_BYTES + 64;
    float* mag = (float*)(ws + GRAY_BYTES);
    float* acc = have_mag ? (float*)(ws + 2 * GRAY_BYTES)
                          : (float*)(ws + GRAY_BYTES);

    hipMemsetAsync(acc, 0, 4 * sizeof(float), stream);          // capture-safe

    k_pass1<<<RED_BLOCKS, BLOCK, 0, stream>>>(pred, tgt, gray, acc);
    if (have_mag) {
        const dim3 tiles(IMG_W / TS, IMG_H / TS, IMG_B);        // 16x16x16
        k_mag <<<tiles, BLOCK, 0, stream>>>(gray, mag);
        k_edge<<<tiles, BLOCK, 0, stream>>>(mag, acc);
    } else {
        k_edge_otf<<<RED_BLOCKS, BLOCK, 0, stream>>>(gray, acc);
    }
    k_final<<<1, 64, 0, stream>>>(acc, out);
}